// DistributionNMS_12008728559697
// MI455X (gfx1250) — compile-verified
//
#include <hip/hip_runtime.h>
#include <hip/hip_bf16.h>
#include <math.h>

// ---------------------------------------------------------------------------
// DistributionNMS for MI455X (gfx1250, wave32)
// B=8, N=2048, C=80, MAX_DET=512, IOU_TH=0.5, SCORE_TH=0.5
// ---------------------------------------------------------------------------

#define BB   8
#define NN   2048
#define CC   80
#define MAXD 512

typedef float v2f __attribute__((ext_vector_type(2)));
typedef float v8f __attribute__((ext_vector_type(8)));

// ---------------------------------------------------------------------------
// Kernel 1: per-box class max -> conf = sigmoid(max); write unnormalized
// exp(x - max) into probs workspace.
// ---------------------------------------------------------------------------
__global__ void nms_prep_kernel(const float* __restrict__ cls,
                                float* __restrict__ probs,
                                float* __restrict__ conf,
                                int total /* B*N */) {
    int n = blockIdx.x * blockDim.x + threadIdx.x;
    if (n >= total) return;
    const float4* r4 = (const float4*)(cls + (size_t)n * CC);
    float m = -INFINITY;
    #pragma unroll
    for (int i = 0; i < CC / 4; ++i) {
        float4 t = r4[i];
        m = fmaxf(m, fmaxf(fmaxf(t.x, t.y), fmaxf(t.z, t.w)));
    }
    float4* o4 = (float4*)(probs + (size_t)n * CC);
    #pragma unroll
    for (int i = 0; i < CC / 4; ++i) {
        float4 t = r4[i];
        float4 e;
        e.x = __expf(t.x - m);
        e.y = __expf(t.y - m);
        e.z = __expf(t.z - m);
        e.w = __expf(t.w - m);
        o4[i] = e;
    }
    conf[n] = 1.0f / (1.0f + __expf(-m));
}

// ---------------------------------------------------------------------------
// Kernel 2: softmax denominator via V_WMMA_F32_16X16X4_F32.
// One wave (32 threads) per 16 rows. A = exp tile [16 x 4] per step,
// B = ones [4 x 16] => D[m][n] = rowsum(m) broadcast across columns.
// A-layout (ISA 16x4 f32): lanes 0-15 hold (K=k0,k0+1) of row=lane,
// lanes 16-31 hold (K=k0+2,k0+3) of row=lane-16.
// D-layout: lane<16: d[j] = row j;  lane>=16: d[j] = row j+8.
// ---------------------------------------------------------------------------
__global__ void nms_softmax_wmma_kernel(float* __restrict__ probs) {
    const int lane = threadIdx.x;           // 0..31, one wave per block
    const int row0 = blockIdx.x * 16;       // tile of 16 rows
    const int half = lane >> 4;             // 0 or 1
    const int r    = row0 + (lane & 15);

    v8f acc = {};
    v2f bones;
    bones.x = 1.0f; bones.y = 1.0f;

    #pragma unroll
    for (int k0 = 0; k0 < CC; k0 += 4) {
        const float* p = probs + (size_t)r * CC + k0 + half * 2;
        v2f a;
        a.x = p[0];
        a.y = p[1];
        acc = __builtin_amdgcn_wmma_f32_16x16x4_f32(
            /*neg_a=*/false, a, /*neg_b=*/false, bones,
            /*c_mod=*/(short)0, acc, /*reuse_a=*/false, /*reuse_b=*/false);
    }

    __shared__ float s_sum[16];
    if ((lane & 15) == 0) {                 // lanes 0 and 16
        #pragma unroll
        for (int j = 0; j < 8; ++j) s_sum[half * 8 + j] = acc[j];
    }
    __syncthreads();

    for (int i = lane; i < 16 * CC; i += 32) {
        int rr = i / CC, c = i - rr * CC;
        float inv = 1.0f / s_sum[rr];
        probs[(size_t)(row0 + rr) * CC + c] *= inv;
    }
}

// ---------------------------------------------------------------------------
// Kernel 3: per-image bitonic sort of (conf desc, index). 1024 threads,
// 2048 elements in LDS.
// ---------------------------------------------------------------------------
__global__ void nms_sort_kernel(const float* __restrict__ conf,
                                float* __restrict__ sConf,
                                int* __restrict__ sIdx) {
    const int b = blockIdx.x;
    const int tid = threadIdx.x;            // 0..1023
    __shared__ float key[NN];
    __shared__ int   ind[NN];

    for (int i = tid; i < NN; i += 1024) {
        key[i] = conf[b * NN + i];
        ind[i] = i;
    }
    __syncthreads();

    for (int k = 2; k <= NN; k <<= 1) {
        for (int j = k >> 1; j > 0; j >>= 1) {
            int i = ((tid & ~(j - 1)) << 1) | (tid & (j - 1));
            int p = i + j;
            float a = key[i], c = key[p];
            bool descBlock = ((i & k) == 0);
            bool sw = descBlock ? (a < c) : (a > c);
            if (sw) {
                key[i] = c; key[p] = a;
                int t = ind[i]; ind[i] = ind[p]; ind[p] = t;
            }
            __syncthreads();
        }
    }
    for (int i = tid; i < NN; i += 1024) {
        sConf[b * NN + i] = key[i];
        sIdx [b * NN + i] = ind[i];
    }
}

// ---------------------------------------------------------------------------
// Kernel 4: LDS-resident greedy NMS + compaction + padded output.
// One workgroup (1024 threads) per image. ~50KB LDS.
// ---------------------------------------------------------------------------
__global__ void nms_greedy_kernel(const float* __restrict__ boxes,
                                  const float* __restrict__ probs,
                                  const float* __restrict__ sConf,
                                  const int* __restrict__ sIdx,
                                  float* __restrict__ outBox,
                                  float* __restrict__ outCls,
                                  float* __restrict__ outConf) {
    const int b = blockIdx.x;
    const int tid = threadIdx.x;            // 0..1023

    __shared__ float sy1[NN], sx1[NN], sy2[NN], sx2[NN], scf[NN];
    __shared__ int   soi[NN];
    __shared__ unsigned char skeep[NN];
    __shared__ int   slist[MAXD];
    __shared__ int   sNKeep;

    for (int i = tid; i < NN; i += 1024) {
        int o = sIdx[b * NN + i];
        const float* bp = boxes + ((size_t)b * NN + o) * 4;
        sy1[i] = bp[0]; sx1[i] = bp[1]; sy2[i] = bp[2]; sx2[i] = bp[3];
        float c = sConf[b * NN + i];
        scf[i] = c;
        soi[i] = o;
        skeep[i] = (c > 0.5f) ? (unsigned char)1 : (unsigned char)0;
    }
    __syncthreads();

    // Greedy suppression in descending-score order.
    for (int i = 0; i < NN - 1; ++i) {
        if (scf[i] <= 0.5f) break;          // sorted: remainder can't suppress
        if (skeep[i]) {
            float iy1 = sy1[i], ix1 = sx1[i], iy2 = sy2[i], ix2 = sx2[i];
            float iarea = (iy2 - iy1) * (ix2 - ix1);
            for (int j = i + 1 + tid; j < NN; j += 1024) {
                if (!skeep[j]) continue;
                float ih = fmaxf(fminf(iy2, sy2[j]) - fmaxf(iy1, sy1[j]), 0.0f);
                float iw = fmaxf(fminf(ix2, sx2[j]) - fmaxf(ix1, sx1[j]), 0.0f);
                float inter = ih * iw;
                float jarea = (sy2[j] - sy1[j]) * (sx2[j] - sx1[j]);
                float uni = fmaxf(iarea + jarea - inter, 1e-9f);
                if (inter > 0.5f * uni) skeep[j] = 0;   // IoU > 0.5
            }
        }
        __syncthreads();
    }
    __syncthreads();

    // Compact kept entries (already in descending score order).
    if (tid == 0) {
        int c = 0;
        for (int i = 0; i < NN && c < MAXD; ++i)
            if (skeep[i]) slist[c++] = i;
        sNKeep = c;
    }
    __syncthreads();
    const int nk = sNKeep;

    // conf [512]
    float* ocf = outConf + b * MAXD;
    for (int k = tid; k < MAXD; k += 1024)
        ocf[k] = (k < nk) ? scf[slist[k]] : 0.0f;

    // boxes [512,4]
    float* obx = outBox + (size_t)b * MAXD * 4;
    for (int e = tid; e < MAXD * 4; e += 1024) {
        int k = e >> 2, c = e & 3;
        float v = 0.0f;
        if (k < nk) {
            int i = slist[k];
            v = (c == 0) ? sy1[i] : (c == 1) ? sx1[i] : (c == 2) ? sy2[i] : sx2[i];
        }
        obx[e] = v;
    }

    // cls [512,80]
    float* ocl = outCls + (size_t)b * MAXD * CC;
    for (int e = tid; e < MAXD * CC; e += 1024) {
        int k = e / CC, c = e - k * CC;
        float v = 0.0f;
        if (k < nk) {
            int o = soi[slist[k]];
            v = probs[((size_t)b * NN + o) * CC + c];
        }
        ocl[e] = v;
    }
}

// ---------------------------------------------------------------------------
// Launch
// ---------------------------------------------------------------------------
extern "C" void kernel_launch(void* const* d_in, const int* in_sizes, int n_in,
                              void* d_out, int out_size, void* d_ws, size_t ws_size,
                              hipStream_t stream) {
    const float* boxes = (const float*)d_in[0];   // [8,2048,4]
    const float* cls   = (const float*)d_in[1];   // [8,2048,80]

    // Workspace carve-out (256B aligned chunks)
    char* ws = (char*)d_ws;
    float* probs = (float*)ws;                              // B*N*C floats
    ws += (size_t)BB * NN * CC * sizeof(float);
    float* conf  = (float*)ws;                              // B*N
    ws += (size_t)BB * NN * sizeof(float);
    float* sConf = (float*)ws;                              // B*N
    ws += (size_t)BB * NN * sizeof(float);
    int*   sIdx  = (int*)ws;                                // B*N

    // Output layout: box [8,512,4] | cls [8,512,80] | conf [8,512]
    float* outBox  = (float*)d_out;
    float* outCls  = outBox + (size_t)BB * MAXD * 4;
    float* outConf = outCls + (size_t)BB * MAXD * CC;

    const int total = BB * NN;

    nms_prep_kernel<<<(total + 255) / 256, 256, 0, stream>>>(cls, probs, conf, total);
    nms_softmax_wmma_kernel<<<total / 16, 32, 0, stream>>>(probs);
    nms_sort_kernel<<<BB, 1024, 0, stream>>>(conf, sConf, sIdx);
    nms_greedy_kernel<<<BB, 1024, 0, stream>>>(boxes, probs, sConf, sIdx,
                                               outBox, outCls, outConf);
}